// ImprovedGAT_44822278701845
// MI455X (gfx1250) — compile-verified
//
#include <hip/hip_runtime.h>
#include <math.h>

#define N_NODES 50000
#define N_EDGES 400000
#define E_TOT   (N_EDGES + N_NODES)   // self-loops appended
#define IN_CH   165
#define HEADS   8
#define HID     64
#define FEAT    (HEADS * HID)         // 512
#define NEG_SLOPE 0.2f
#define LN_EPS  1e-5f

typedef __attribute__((ext_vector_type(2))) float v2f;
typedef __attribute__((ext_vector_type(8))) float v8f;

// ---------------------------------------------------------------------------
// GEMM: C[M,N] = A[M,K] @ B[K,N], f32, WMMA 16x16x4 (exact f32 accumulate).
// Block = 256 threads (8 waves). Block tile 32x128; wave tile 16x32
// (two accumulators sharing one A fragment).
// B staged K-pair-interleaved so each B fragment is ONE ds_load_b64.
// ---------------------------------------------------------------------------
#define TM 32
#define TN 128
#define TK 32
#define LDA 40    // As[k][m]: half-wave K-phase offset 2*40 == 16 mod 64 banks
#define TNP 144   // Bs2 row pad: k-pair row stride 2*144 == 32 mod 64 banks

__global__ __launch_bounds__(256)
void gat_gemm_wmma(const float* __restrict__ A, const float* __restrict__ B,
                   float* __restrict__ C, int M, int N, int K) {
  __shared__ float As[TK][LDA];
  __shared__ float Bs2[TK / 2][TNP][2];   // [k/2][n][k&1]

  const int tid  = threadIdx.x;
  const int lane = tid & 31;
  const int wv   = tid >> 5;      // 0..7
  const int wm   = wv >> 2;       // 0..1 (M direction, 16 rows each)
  const int wn   = wv & 3;        // 0..3 (N direction, 32 cols each)
  const int half = lane >> 4;     // K-phase per ISA 16x4 A layout
  const int lm   = lane & 15;

  const int m0 = blockIdx.y * TM;
  const int n0 = blockIdx.x * TN;
  const bool mfull = (m0 + TM) <= M;   // uniform per block

  v8f acc0 = {0.f, 0.f, 0.f, 0.f, 0.f, 0.f, 0.f, 0.f};
  v8f acc1 = {0.f, 0.f, 0.f, 0.f, 0.f, 0.f, 0.f, 0.f};

  for (int kk = 0; kk < K; kk += TK) {
    const bool kfull = (kk + TK) <= K;
    if (mfull && kfull) {
      // fast path: no bounds checks, fully coalesced
#pragma unroll
      for (int i = 0; i < (TM * TK) / 256; ++i) {
        int e = tid + i * 256;
        int mm = e >> 5, kt = e & 31;
        As[kt][mm] = A[(size_t)(m0 + mm) * K + kk + kt];
      }
#pragma unroll
      for (int i = 0; i < (TK * TN) / 256; ++i) {
        int e = tid + i * 256;
        int kt = e >> 7, nn = e & 127;
        Bs2[kt >> 1][nn][kt & 1] = B[(size_t)(kk + kt) * N + n0 + nn];
      }
    } else {
      // tail path: zero-pad out-of-range elements
#pragma unroll
      for (int i = 0; i < (TM * TK) / 256; ++i) {
        int e = tid + i * 256;
        int mm = e >> 5, kt = e & 31;
        int gm = m0 + mm, gk = kk + kt;
        float v = 0.f;
        if (gm < M && gk < K) v = A[(size_t)gm * K + gk];
        As[kt][mm] = v;
      }
#pragma unroll
      for (int i = 0; i < (TK * TN) / 256; ++i) {
        int e = tid + i * 256;
        int kt = e >> 7, nn = e & 127;
        int gk = kk + kt;
        float v = 0.f;
        if (gk < K) v = B[(size_t)gk * N + n0 + nn];   // n0+nn < N always
        Bs2[kt >> 1][nn][kt & 1] = v;
      }
    }
    // prefetch next K tile with clamped (non-divergent) addresses
    if (kk + TK < K) {
      int mm = tid >> 5, kt = tid & 31;
      int gm = m0 + mm; if (gm >= M) gm = M - 1;
      int gk = kk + TK + kt; if (gk >= K) gk = K - 1;
      __builtin_prefetch(&A[(size_t)gm * K + gk], 0, 1);
      int kb = tid >> 7, nn = tid & 127;
      int gkb = kk + TK + kb; if (gkb >= K) gkb = K - 1;
      __builtin_prefetch(&B[(size_t)gkb * N + n0 + nn], 0, 1);
    }
    __syncthreads();

#pragma unroll
    for (int k = 0; k < TK; k += 4) {
      int ka = k + half * 2;        // A K-phase row
      int kp = (k >> 1) + half;     // B k-pair row
      v2f a, bf0, bf1;
      a.x = As[ka][wm * 16 + lm];
      a.y = As[ka + 1][wm * 16 + lm];
      bf0 = *(const v2f*)(&Bs2[kp][wn * 32 + lm][0]);
      bf1 = *(const v2f*)(&Bs2[kp][wn * 32 + 16 + lm][0]);
#if __has_builtin(__builtin_amdgcn_wmma_f32_16x16x4_f32)
      acc0 = __builtin_amdgcn_wmma_f32_16x16x4_f32(
          false, a, false, bf0, (short)0, acc0, false, false);
      acc1 = __builtin_amdgcn_wmma_f32_16x16x4_f32(
          false, a, false, bf1, (short)0, acc1, false, false);
#else
      for (int r = 0; r < 8; ++r) {
        acc0[r] += a.x * bf0.x + a.y * bf0.y;
        acc1[r] += a.x * bf1.x + a.y * bf1.y;
      }
#endif
    }
    __syncthreads();
  }

  // C/D layout: VGPR r -> M = r (+8 for upper half-wave), N = lane mod 16
#pragma unroll
  for (int r = 0; r < 8; ++r) {
    int gm = m0 + wm * 16 + half * 8 + r;
    int gn = n0 + wn * 32 + lm;
    if (gm < M) {
      C[(size_t)gm * N + gn] = acc0[r];
      C[(size_t)gm * N + gn + 16] = acc1[r];
    }
  }
}

// ---------------------------------------------------------------------------
// Attention logits: alpha_s[n,h] = dot(h[n,h,:], a_src[h,:]); same for dst.
// One block (8 waves) per node; one wave per head; wave-shuffle reduce.
// ---------------------------------------------------------------------------
__global__ __launch_bounds__(256)
void gat_alpha(const float* __restrict__ h, const float* __restrict__ aw_s,
               const float* __restrict__ aw_d, float* __restrict__ as_o,
               float* __restrict__ ad_o) {
  int n = blockIdx.x;
  int hd = threadIdx.x >> 5;
  int lane = threadIdx.x & 31;
  const float* hp = h + (size_t)n * FEAT + hd * HID;
  float x0 = hp[lane], x1 = hp[lane + 32];
  float s = x0 * aw_s[hd * HID + lane] + x1 * aw_s[hd * HID + lane + 32];
  float d = x0 * aw_d[hd * HID + lane] + x1 * aw_d[hd * HID + lane + 32];
  for (int off = 16; off; off >>= 1) {
    s += __shfl_down(s, off);
    d += __shfl_down(d, off);
  }
  if (lane == 0) {
    as_o[n * HEADS + hd] = s;
    ad_o[n * HEADS + hd] = d;
  }
}

// order-preserving float<->uint map for atomic max over signed floats
__device__ __forceinline__ unsigned f32_order(float f) {
  unsigned u = __float_as_uint(f);
  return (u & 0x80000000u) ? ~u : (u | 0x80000000u);
}
__device__ __forceinline__ float f32_unorder(unsigned u) {
  return (u & 0x80000000u) ? __uint_as_float(u & 0x7fffffffu)
                           : __uint_as_float(~u);
}

// pass 1: e = leaky_relu(as[src]+ad[dst]); segment max via atomicMax
__global__ void gat_edge_max(const int* __restrict__ ei,
                             const float* __restrict__ as_o,
                             const float* __restrict__ ad_o,
                             float* __restrict__ ebuf,
                             unsigned* __restrict__ mx) {
  int idx = blockIdx.x * blockDim.x + threadIdx.x;
  if (idx >= E_TOT * HEADS) return;
  int e = idx >> 3, hd = idx & 7;
  int s, d;
  if (e < N_EDGES) { s = ei[e]; d = ei[N_EDGES + e]; }
  else             { s = d = e - N_EDGES; }
  float v = as_o[s * HEADS + hd] + ad_o[d * HEADS + hd];
  v = (v > 0.f) ? v : NEG_SLOPE * v;
  ebuf[(size_t)e * HEADS + hd] = v;
  atomicMax(&mx[d * HEADS + hd], f32_order(v));
}

// pass 2: ex = exp(e - m[dst]); denom[dst] += ex
__global__ void gat_edge_exp(const int* __restrict__ ei,
                             float* __restrict__ ebuf,
                             const unsigned* __restrict__ mx,
                             float* __restrict__ denom) {
  int idx = blockIdx.x * blockDim.x + threadIdx.x;
  if (idx >= E_TOT * HEADS) return;
  int e = idx >> 3, hd = idx & 7;
  int d = (e < N_EDGES) ? ei[N_EDGES + e] : (e - N_EDGES);
  float m = f32_unorder(mx[d * HEADS + hd]);
  float ex = __expf(ebuf[(size_t)e * HEADS + hd] - m);
  ebuf[(size_t)e * HEADS + hd] = ex;
  atomicAdd(&denom[d * HEADS + hd], ex);
}

// pass 3: agg[dst] += (ex/denom[dst]) * h[src]   (one 256-thread block / edge)
__global__ __launch_bounds__(256)
void gat_edge_agg(const int* __restrict__ ei, const float* __restrict__ h,
                  const float* __restrict__ ebuf,
                  const float* __restrict__ denom, float* __restrict__ agg) {
  int e = blockIdx.x;
  int s, d;
  if (e < N_EDGES) { s = ei[e]; d = ei[N_EDGES + e]; }
  else             { s = d = e - N_EDGES; }
  __shared__ float coeff[HEADS];
  if (threadIdx.x < HEADS)
    coeff[threadIdx.x] =
        ebuf[(size_t)e * HEADS + threadIdx.x] / denom[d * HEADS + threadIdx.x];
  __syncthreads();
  const float* hs = h + (size_t)s * FEAT;
  float* od = agg + (size_t)d * FEAT;
  int t = threadIdx.x;
  atomicAdd(&od[t], coeff[t >> 6] * hs[t]);
  atomicAdd(&od[t + 256], coeff[(t + 256) >> 6] * hs[t + 256]);
}

// bias (+ optional residual) + LayerNorm(512) + ReLU, in place
__global__ __launch_bounds__(256)
void gat_epilogue512(float* __restrict__ agg, const float* __restrict__ bias,
                     const float* __restrict__ resid,
                     const float* __restrict__ g, const float* __restrict__ b) {
  __shared__ float red[8];
  int n = blockIdx.x, t = threadIdx.x;
  size_t base = (size_t)n * FEAT;
  float v0 = agg[base + t] + bias[t];
  float v1 = agg[base + t + 256] + bias[t + 256];
  if (resid) { v0 += resid[base + t]; v1 += resid[base + t + 256]; }
  int lane = t & 31, wv = t >> 5;
  float s = v0 + v1;
  for (int off = 16; off; off >>= 1) s += __shfl_down(s, off);
  if (lane == 0) red[wv] = s;
  __syncthreads();
  if (t == 0) { float tot = 0; for (int i = 0; i < 8; ++i) tot += red[i]; red[0] = tot; }
  __syncthreads();
  float mu = red[0] * (1.0f / FEAT);
  __syncthreads();
  float d0 = v0 - mu, d1 = v1 - mu;
  s = d0 * d0 + d1 * d1;
  for (int off = 16; off; off >>= 1) s += __shfl_down(s, off);
  if (lane == 0) red[wv] = s;
  __syncthreads();
  if (t == 0) { float tot = 0; for (int i = 0; i < 8; ++i) tot += red[i]; red[0] = tot; }
  __syncthreads();
  float rstd = rsqrtf(red[0] * (1.0f / FEAT) + LN_EPS);
  agg[base + t] = fmaxf(d0 * rstd * g[t] + b[t], 0.f);
  agg[base + t + 256] = fmaxf(d1 * rstd * g[t + 256] + b[t + 256], 0.f);
}

// layer 2: mean over heads -> [N,64], +bias, LayerNorm(64), ReLU
__global__ __launch_bounds__(64)
void gat_epilogue64(const float* __restrict__ agg, const float* __restrict__ bias,
                    const float* __restrict__ g, const float* __restrict__ b,
                    float* __restrict__ out64) {
  __shared__ float red[2];
  int n = blockIdx.x, t = threadIdx.x;
  size_t base = (size_t)n * FEAT;
  float v = 0.f;
#pragma unroll
  for (int hd = 0; hd < HEADS; ++hd) v += agg[base + hd * HID + t];
  v = v * (1.0f / HEADS) + bias[t];
  int lane = t & 31, wv = t >> 5;
  float s = v;
  for (int off = 16; off; off >>= 1) s += __shfl_down(s, off);
  if (lane == 0) red[wv] = s;
  __syncthreads();
  float mu = (red[0] + red[1]) * (1.0f / HID);
  __syncthreads();
  float dv = v - mu;
  s = dv * dv;
  for (int off = 16; off; off >>= 1) s += __shfl_down(s, off);
  if (lane == 0) red[wv] = s;
  __syncthreads();
  float rstd = rsqrtf((red[0] + red[1]) * (1.0f / HID) + LN_EPS);
  out64[(size_t)n * HID + t] = fmaxf(dv * rstd * g[t] + b[t], 0.f);
}

// out[N,2] = f64 @ lin_w + lin_b
__global__ void gat_final(const float* __restrict__ f64,
                          const float* __restrict__ w,
                          const float* __restrict__ b, float* __restrict__ out) {
  int i = blockIdx.x * blockDim.x + threadIdx.x;
  if (i >= N_NODES) return;
  float a0 = b[0], a1 = b[1];
  const float* f = f64 + (size_t)i * HID;
#pragma unroll
  for (int c = 0; c < HID; ++c) {
    a0 += f[c] * w[2 * c];
    a1 += f[c] * w[2 * c + 1];
  }
  out[2 * i] = a0;
  out[2 * i + 1] = a1;
}

// ---------------------------------------------------------------------------
extern "C" void kernel_launch(void* const* d_in, const int* in_sizes, int n_in,
                              void* d_out, int out_size, void* d_ws,
                              size_t ws_size, hipStream_t stream) {
  (void)in_sizes; (void)n_in; (void)out_size; (void)ws_size;
  const float* x    = (const float*)d_in[0];
  const int*   ei   = (const int*)d_in[1];
  const float* W0   = (const float*)d_in[2];
  const float* as0  = (const float*)d_in[3];
  const float* ad0  = (const float*)d_in[4];
  const float* b0   = (const float*)d_in[5];
  const float* lng0 = (const float*)d_in[6];
  const float* lnb0 = (const float*)d_in[7];
  const float* W1   = (const float*)d_in[8];
  const float* as1  = (const float*)d_in[9];
  const float* ad1  = (const float*)d_in[10];
  const float* b1   = (const float*)d_in[11];
  const float* lng1 = (const float*)d_in[12];
  const float* lnb1 = (const float*)d_in[13];
  const float* W2   = (const float*)d_in[14];
  const float* as2  = (const float*)d_in[15];
  const float* ad2  = (const float*)d_in[16];
  const float* b2   = (const float*)d_in[17];
  const float* lng2 = (const float*)d_in[18];
  const float* lnb2 = (const float*)d_in[19];
  const float* lw   = (const float*)d_in[20];
  const float* lb   = (const float*)d_in[21];

  char* ws = (char*)d_ws;
  size_t off = 0;
  auto alloc = [&](size_t bytes) -> void* {
    void* p = ws + off;
    off += (bytes + 255) & ~(size_t)255;
    return p;
  };
  const size_t big = (size_t)N_NODES * FEAT * sizeof(float);
  float*    B0    = (float*)alloc(big);
  float*    B1    = (float*)alloc(big);
  float*    B2    = (float*)alloc(big);
  float*    as_o  = (float*)alloc((size_t)N_NODES * HEADS * 4);
  float*    ad_o  = (float*)alloc((size_t)N_NODES * HEADS * 4);
  unsigned* mx    = (unsigned*)alloc((size_t)N_NODES * HEADS * 4);
  float*    denom = (float*)alloc((size_t)N_NODES * HEADS * 4);
  float*    ebuf  = (float*)alloc((size_t)E_TOT * HEADS * 4);
  float*    f64   = (float*)alloc((size_t)N_NODES * HID * 4);

  dim3 ggrid((FEAT + TN - 1) / TN, (N_NODES + TM - 1) / TM);
  int nt = E_TOT * HEADS;
  int eb = (nt + 255) / 256;

  auto run_layer = [&](const float* in, int K, const float* W,
                       const float* aws, const float* awd, float* hbuf,
                       float* aggbuf) {
    gat_gemm_wmma<<<ggrid, 256, 0, stream>>>(in, W, hbuf, N_NODES, FEAT, K);
    gat_alpha<<<N_NODES, 256, 0, stream>>>(hbuf, aws, awd, as_o, ad_o);
    hipMemsetAsync(mx, 0, (size_t)N_NODES * HEADS * 4, stream);
    hipMemsetAsync(denom, 0, (size_t)N_NODES * HEADS * 4, stream);
    hipMemsetAsync(aggbuf, 0, big, stream);
    gat_edge_max<<<eb, 256, 0, stream>>>(ei, as_o, ad_o, ebuf, mx);
    gat_edge_exp<<<eb, 256, 0, stream>>>(ei, ebuf, mx, denom);
    gat_edge_agg<<<E_TOT, 256, 0, stream>>>(ei, hbuf, ebuf, denom, aggbuf);
  };

  // layer 0: x[50000,165] -> B1 (no residual: shapes differ)
  run_layer(x, IN_CH, W0, as0, ad0, B0, B1);
  gat_epilogue512<<<N_NODES, 256, 0, stream>>>(B1, b0, nullptr, lng0, lnb0);

  // layer 1: B1 -> B2 (+residual B1)
  run_layer(B1, FEAT, W1, as1, ad1, B0, B2);
  gat_epilogue512<<<N_NODES, 256, 0, stream>>>(B2, b1, B1, lng1, lnb1);

  // layer 2: B2 -> B1, head-mean + LN(64) -> f64
  run_layer(B2, FEAT, W2, as2, ad2, B0, B1);
  gat_epilogue64<<<N_NODES, 64, 0, stream>>>(B1, b2, lng2, lnb2, f64);

  gat_final<<<(N_NODES + 255) / 256, 256, 0, stream>>>(f64, lw, lb,
                                                       (float*)d_out);
}